// DifferentiableVolumeRendering_27891517620641
// MI455X (gfx1250) — compile-verified
//
#include <hip/hip_runtime.h>
#include <math.h>

// DifferentiableVolumeRendering on gfx1250 (MI455X).
// One ray per lane (wave32), 192 streaming steps, running-transmittance form:
//   T *= exp(-sigma*delta);  pred += T_before * (1 - exp(-sigma*delta)) * t_k
// Memory-latency bound (12.6M gathers into a 201MB volume, ~fits 192MB L2):
//  - 2048 waves for occupancy-based latency hiding
//  - unroll-8 so 8 independent gathers are in flight per wave
//  - gfx1250 global_prefetch_b8 issued 32 steps (~1.3k cycles) ahead per group
//  - branch-free clamped gather (v_med3 + v_cndmask, no exec-mask churn)

namespace {
constexpr int   kT     = 6;
constexpr int   kZ     = 32;
constexpr int   kY     = 512;   // 2^9
constexpr int   kX     = 512;   // 2^9  -> flat index = (iz<<18)|(iy<<9)|ix
constexpr int   kSteps = 192;
constexpr float kVoxel = 0.2f;
constexpr int   kPF    = 32;    // prefetch distance in steps
}

__global__ __launch_bounds__(256)
void volrender_kernel(const float* __restrict__ sigma,   // [T,Z,Y,X] (B=1)
                      const float* __restrict__ org,     // [T,3]
                      const float* __restrict__ pcd,     // [N,3]
                      const int*   __restrict__ tindex,  // [N]
                      float*       __restrict__ out,     // [2*N]: pred | gt
                      int N)
{
  const int n = blockIdx.x * blockDim.x + threadIdx.x;
  if (n >= N) return;

  int t = tindex[n];
  t = (t < 0) ? 0 : (t >= kT ? kT - 1 : t);

  const float ox = org[3 * t + 0], oy = org[3 * t + 1], oz = org[3 * t + 2];
  const float ex = pcd[3 * n + 0], ey = pcd[3 * n + 1], ez = pcd[3 * n + 2];

  const float rx = ex - ox, ry = ey - oy, rz = ez - oz;
  const float gt = sqrtf(fmaf(rx, rx, fmaf(ry, ry, rz * rz)));

  constexpr float inv = 1.0f / (float)kSteps;
  const float sx = rx * inv, sy = ry * inv, sz = rz * inv;  // per-step advance
  const float delta = gt * inv;                              // step length

  const float* __restrict__ base =
      sigma + (size_t)t * ((size_t)kZ * kY * kX);

  // midpoint sampling: pos_k = org + ray*(k+0.5)/S ; t_k = delta*(k+0.5)
  float fx = fmaf(sx, 0.5f, ox);
  float fy = fmaf(sy, 0.5f, oy);
  float fz = fmaf(sz, 0.5f, oz);
  float tk = 0.5f * delta;

  float trans = 1.0f;   // transmittance entering step k
  float pred  = 0.0f;

  #pragma unroll 8
  for (int k = 0; k < kSteps; ++k) {
    // gfx1250 prefetch path: one global_prefetch_b8 per unrolled group,
    // targeting the voxel the ray will gather kPF steps from now (RT hint:
    // the volume ~fits in the 192MB L2 and lines are re-used across rays).
    if ((k & 7) == 0) {
      const float qx = fmaf(sx, (float)kPF, fx);
      const float qy = fmaf(sy, (float)kPF, fy);
      const float qz = fmaf(sz, (float)kPF, fz);
      int jx = (int)floorf(qx); jx = jx < 0 ? 0 : (jx > kX - 1 ? kX - 1 : jx);
      int jy = (int)floorf(qy); jy = jy < 0 ? 0 : (jy > kY - 1 ? kY - 1 : jy);
      int jz = (int)floorf(qz); jz = jz < 0 ? 0 : (jz > kZ - 1 ? kZ - 1 : jz);
      __builtin_prefetch(
          base + (((size_t)jz << 18) | ((size_t)jy << 9) | (size_t)jx), 0, 3);
    }

    const int ix = (int)floorf(fx);
    const int iy = (int)floorf(fy);
    const int iz = (int)floorf(fz);

    const bool inside = ((unsigned)ix < (unsigned)kX) &
                        ((unsigned)iy < (unsigned)kY) &
                        ((unsigned)iz < (unsigned)kZ);

    // Branch-free clamped gather (v_med3_i32), then select 0 if outside.
    const int cx = ix < 0 ? 0 : (ix > kX - 1 ? kX - 1 : ix);
    const int cy = iy < 0 ? 0 : (iy > kY - 1 ? kY - 1 : iy);
    const int cz = iz < 0 ? 0 : (iz > kZ - 1 ? kZ - 1 : iz);
    const float sv = base[((size_t)cz << 18) | ((size_t)cy << 9) | (size_t)cx];
    const float s  = inside ? sv : 0.0f;

    // tau = s*delta; alpha = 1-exp(-tau); w = trans*alpha; pred += w*t_k
    const float e = __expf(-s * delta);          // single v_exp per step
    pred = fmaf(trans * (1.0f - e), tk, pred);
    trans *= e;

    fx += sx; fy += sy; fz += sz; tk += delta;
  }

  // residual transmittance terminates at ray end
  pred = fmaf(trans, gt, pred);

  // NaN/Inf cleanup as in the reference wrapper (gt masked by pred's badness)
  const bool bad = __builtin_isnan(pred) || __builtin_isinf(pred);
  const float predc = bad ? 0.0f : pred;
  const float gtc   = bad ? 0.0f : gt;

  out[n]     = predc * kVoxel;
  out[N + n] = gtc   * kVoxel;
}

extern "C" void kernel_launch(void* const* d_in, const int* in_sizes, int n_in,
                              void* d_out, int out_size, void* d_ws, size_t ws_size,
                              hipStream_t stream) {
  const float* sigma  = (const float*)d_in[0];  // [1,6,32,512,512] f32
  const float* org    = (const float*)d_in[1];  // [1,6,3] f32
  const float* pcd    = (const float*)d_in[2];  // [1,65536,3] f32
  const int*   tindex = (const int*)d_in[3];    // [1,65536] i32
  float* out = (float*)d_out;                   // [2,65536] f32 (pred, gt)

  const int N = in_sizes[3];                    // 65536 rays
  const int block = 256;                        // 8 wave32 waves per block
  const int grid  = (N + block - 1) / block;    // 2048 waves total
  volrender_kernel<<<grid, block, 0, stream>>>(sigma, org, pcd, tindex, out, N);
}